// SymmetricContraction_2645699854347
// MI455X (gfx1250) — compile-verified
//
#include <hip/hip_runtime.h>
#include <hip/hip_bf16.h>

// MACE-style symmetric contraction for MI455X (gfx1250, wave32, WMMA).
//
// Factorized algorithm (~1.3e10 FLOP vs ~5e13 naive):
//   w[s,k,c]   = sum_e species_table[s,e] * W[e,k,c]           (tiny GEMMs)
//   V3[s,c,x,y,j,i] = sum_k U3[x,y,j,k,i] * w[s,k,c]           (WMMA build)
//   per node b of species s:
//     M1[x,y,i] = sum_j V3[x,y,j,i]*x_j                         (WMMA apply)
//     M2[x,i]   = sum_y (M1 + V2[x,y,i]) * x_y                  (VALU/LDS)
//     out[i]    = sum_x (M2 + V1[x,i])   * x_x                  (VALU/LDS)
// Channel-tiled into 8 passes of 16 channels so V3 per pass = 64 MB < 192 MB L2.
//
// This revision removes all per-lane K-guards (zero-padded packed operands) so
// every WMMA A/B fragment is an unconditional contiguous global_load_b64, and
// the V3 tile store is 2x global_store_b128 per lane.

typedef __attribute__((ext_vector_type(2))) float v2f;
typedef __attribute__((ext_vector_type(4))) float v4f;
typedef __attribute__((ext_vector_type(8))) float v8f;

#define NN 1024
#define CC 128
#define DD 16
#define SS 64
#define EE 32
#define KP 36   // unified padded K for order-3 (covers 23 and 33)

// ---------------------------------------------------------------------------
// Kernel 1: per-species weights.
//  w3t0[s][c][36] / w3t1[s][c][36]: transposed + zero-padded (B fragments are
//  contiguous v2f loads).  w2/w1 stay [s][k][c] (scalar use in apply).
// ---------------------------------------------------------------------------
__global__ __launch_bounds__(256)
void prep_weights(const float* st,
                  const float* W3_0e, const float* W3_1o,
                  const float* W2_0e, const float* W2_1o,
                  const float* W1_0e, const float* W1_1o,
                  float* w3t0, float* w3t1, float* w2_0e, float* w2_1o,
                  float* w1_0e, float* w1_1o) {
  int idx = blockIdx.x * blockDim.x + threadIdx.x;
  if (idx < SS*CC*KP || (idx -= SS*CC*KP) < SS*CC*KP) {
    // transposed w3 segments
    const float* W = (idx == blockIdx.x * blockDim.x + threadIdx.x) ? W3_0e : W3_1o;
    float* o       = (W == W3_0e) ? w3t0 : w3t1;
    int K          = (W == W3_0e) ? 23 : 33;
    int s = idx / (CC*KP);
    int c = (idx / KP) % CC;
    int k = idx % KP;
    float v = 0.f;
    if (k < K)
      for (int e = 0; e < EE; ++e)
        v += st[s*EE + e] * W[(e*K + k)*CC + c];
    o[(s*CC + c)*KP + k] = v;
    return;
  }
  const float* W; float* o; int Kp, K;
  if ((idx) < SS*4*CC)                   { W = W2_0e; o = w2_0e; Kp = 4; K = 4; }
  else if ((idx -= SS*4*CC) < SS*6*CC)   { W = W2_1o; o = w2_1o; Kp = 6; K = 6; }
  else if ((idx -= SS*6*CC) < SS*1*CC)   { W = W1_0e; o = w1_0e; Kp = 1; K = 1; }
  else if ((idx -= SS*1*CC) < SS*1*CC)   { W = W1_1o; o = w1_1o; Kp = 1; K = 1; }
  else return;
  int s = idx / (Kp*CC);
  int k = (idx / CC) % Kp;
  int c = idx % CC;
  float v = 0.f;
  for (int e = 0; e < EE; ++e)
    v += st[s*EE + e] * W[(e*K + k)*CC + c];
  o[idx] = v;
}

// ---------------------------------------------------------------------------
// Kernel 1b: pack U3 into WMMA-A-friendly layout UA[m][j][KP], m=(x*16+y)*4+i,
// zero-padded in k.  A fragments become unconditional contiguous v2f loads.
// ---------------------------------------------------------------------------
__global__ __launch_bounds__(256)
void prep_ua(const float* U3_0e, const float* U3_1o, float* UA) {
  int idx = blockIdx.x * blockDim.x + threadIdx.x;
  if (idx >= 1024*DD*KP) return;
  int m  = idx / (DD*KP);
  int j  = (idx / KP) % DD;
  int k  = idx % KP;
  int ii = m & 3;
  int xy = m >> 2;
  float v = 0.f;
  if (ii == 0) { if (k < 23) v = U3_0e[(xy*DD + j)*23 + k]; }
  else         { if (k < 33) v = U3_1o[((xy*DD + j)*33 + k)*3 + (ii - 1)]; }
  UA[idx] = v;
}

// ---------------------------------------------------------------------------
// Kernel 2: bin nodes by species (counting sort; in-bin order is irrelevant
// to the result, so atomic ordering nondeterminism does not affect output).
// ---------------------------------------------------------------------------
__global__ __launch_bounds__(NN)
void bin_nodes(const int* index, int* counts, int* offsets, int* nodelist) {
  __shared__ int cnt[SS];
  __shared__ int base[SS];
  __shared__ int alloc[SS];
  int t = threadIdx.x;
  if (t < SS) cnt[t] = 0;
  __syncthreads();
  int s = index[t] & (SS - 1);
  atomicAdd(&cnt[s], 1);
  __syncthreads();
  if (t == 0) {
    int a = 0;
    for (int i = 0; i < SS; ++i) { base[i] = a; alloc[i] = a; a += cnt[i]; }
  }
  __syncthreads();
  if (t < SS) { counts[t] = cnt[t]; offsets[t] = base[t]; }
  int pos = atomicAdd(&alloc[s], 1);
  nodelist[pos] = t;
}

// ---------------------------------------------------------------------------
// Kernel 3: build V3 with V_WMMA_F32_16X16X4_F32 (uniform 9 K-steps).
// Block = 1 wave32. blockIdx.x = m = ((x*16+y)*4 + i), blockIdx.y = species.
// D-tile rows = j (16), cols = 16 channels of this pass.
// V3 layout: [s][m][c_local][j]  (j fastest -> apply-A loads are v2f;
// D store here is 2x b128 per lane).
// ---------------------------------------------------------------------------
__global__ __launch_bounds__(32)
void build_v3(const float* UA, const float* w3t0, const float* w3t1,
              float* V3, int cbase) {
  int m  = blockIdx.x;
  int s  = blockIdx.y;
  int l  = threadIdx.x;
  int ii = m & 3;
  int col = l & 15;           // A: row j   / B,D: column (channel)
  int hi  = (l >> 4) & 1;     // half-wave -> K pair {0,1} vs {2,3}
  int c = cbase + col;

  const float* Up = UA + ((size_t)m * DD + col) * KP;                // A row
  const float* Wp = (ii == 0 ? w3t0 : w3t1) + ((size_t)s*CC + c)*KP; // B col

  v8f acc = {};
  for (int kb = 0; kb < KP; kb += 4) {
    int k0 = kb + hi*2;
    v2f a = *(const v2f*)(Up + k0);
    v2f b = *(const v2f*)(Wp + k0);
    acc = __builtin_amdgcn_wmma_f32_16x16x4_f32(false, a, false, b,
                                                (short)0, acc, false, false);
  }
  // D layout: VGPR d holds row j = d + hi*8, column = l%16  ->  8 contiguous
  // floats per lane in V3[s][m][col][j]; store as two b128.
  float* o = V3 + (((size_t)s * 1024 + m) * 16 + col) * 16 + hi*8;
  v4f lo = { acc[0], acc[1], acc[2], acc[3] };
  v4f hi4 = { acc[4], acc[5], acc[6], acc[7] };
  *(v4f*)(o)     = lo;
  *(v4f*)(o + 4) = hi4;
}

// ---------------------------------------------------------------------------
// Kernel 4: apply phase. Block = 1 wave32, grid = (16 channels, 64 species).
// Per node-tile of 16 nodes (same species): the j-contraction is a WMMA GEMM
//   D[(y,i), b] = sum_j V3[x,y,j,i] * x[b,c,j]
// then y- and x-contractions are fused via LDS with U2/U1 terms injected.
// ---------------------------------------------------------------------------
__global__ __launch_bounds__(32)
void apply_contract(const float* x, const float* V3,
                    const float* w2_0e, const float* w2_1o,
                    const float* w1_0e, const float* w1_1o,
                    const float* U2_0e, const float* U2_1o,
                    const float* U1_0e, const float* U1_1o,
                    const int* counts, const int* offsets, const int* nodelist,
                    float* out, int cbase) {
  int cl = blockIdx.x;
  int s  = blockIdx.y;
  int l  = threadIdx.x;
  int c  = cbase + cl;
  int cnt = counts[s];
  if (cnt == 0) return;          // uniform per block -> EXEC stays full
  int off = offsets[s];

  __shared__ float V2l[DD*DD*4]; // [x][y][i]
  __shared__ float V1l[DD*4];    // [x][i]
  __shared__ float xs[16*DD];    // [b][d]
  __shared__ float M2[4*16];     // [i][b]

  // V2[x,y,i] = sum_k U2[x,y,k,i] * w2[s,k,c]   (cheap, once per block)
  for (int e = l; e < DD*DD*4; e += 32) {
    int xx = e >> 6, yy = (e >> 2) & 15, i2 = e & 3;
    float v = 0.f;
    if (i2 == 0) {
      for (int k = 0; k < 4; ++k)
        v += U2_0e[(xx*DD + yy)*4 + k] * w2_0e[(s*4 + k)*CC + c];
    } else {
      for (int k = 0; k < 6; ++k)
        v += U2_1o[((xx*DD + yy)*6 + k)*3 + (i2 - 1)] * w2_1o[(s*6 + k)*CC + c];
    }
    V2l[e] = v;
  }
  for (int e = l; e < DD*4; e += 32) {
    int xx = e >> 2, i2 = e & 3;
    V1l[e] = (i2 == 0) ? U1_0e[xx] * w1_0e[s*CC + c]
                       : U1_1o[xx*3 + (i2 - 1)] * w1_1o[s*CC + c];
  }
  __syncthreads();

  int col = l & 15;
  int hi  = (l >> 4) & 1;
  int ntiles = (cnt + 15) >> 4;
  const float* V3s = V3 + (size_t)s * (1024 * 16 * 16);

  for (int nt = 0; nt < ntiles; ++nt) {
    // stage x for 16 nodes (padded slots replicate first node; their columns
    // are computed but never stored)
    for (int e = l; e < 256; e += 32) {
      int b = e >> 4, dd = e & 15;
      int bi = nt*16 + b;
      int node = nodelist[off + (bi < cnt ? bi : 0)];
      xs[b*DD + dd] = x[((size_t)node*CC + c)*DD + dd];
    }
    for (int e = l; e < 64; e += 32) M2[e] = 0.f;
    __syncthreads();

    // B fragments: B[j,b] = xs[b][j]; lane col b = l%16 -> contiguous v2f
    v2f vb[4];
    for (int kb = 0; kb < 4; ++kb)
      vb[kb] = *(const v2f*)&xs[col*DD + kb*4 + hi*2];

    float oacc0 = 0.f, oacc1 = 0.f;  // lane covers entries l and l+32 of [i][b]

    for (int xx = 0; xx < DD; ++xx) {
      for (int t = 0; t < 4; ++t) {
        // A row m = l%16 -> (y,i) = t*16 + m; K = j (contiguous in V3)
        int yiA = t*16 + col;
        int yA = yiA >> 2, iA = yiA & 3;
        const float* Arow = V3s + (((size_t)((xx*DD + yA)*4 + iA)) * 16 + cl) * 16;
        v8f acc = {};
        for (int kb = 0; kb < 4; ++kb) {
          v2f a = *(const v2f*)(Arow + kb*4 + hi*2);
          acc = __builtin_amdgcn_wmma_f32_16x16x4_f32(false, a, false, vb[kb],
                                                      (short)0, acc, false, false);
        }
        // D rows: VGPR d -> (y,i) = t*16 + d + hi*8 ; column b = l%16.
        // M2[i][b] += (M1 + V2[x,y,i]) * x[b,c,y]
        for (int d = 0; d < 8; ++d) {
          int yi = t*16 + d + hi*8;
          int yy = yi >> 2, i3 = yi & 3;
          float val = acc[d] + V2l[(xx*DD + yy)*4 + i3];
          atomicAdd(&M2[i3*16 + col], val * xs[col*DD + yy]);
        }
      }
      __syncthreads();
      // out[i,b] += (M2[i,b] + V1[x,i]) * x[b,c,x]
      {
        int e0 = l, e1 = l + 32;
        int i0 = e0 >> 4, b0 = e0 & 15;
        int i1 = e1 >> 4, b1 = e1 & 15;
        oacc0 += (M2[e0] + V1l[xx*4 + i0]) * xs[b0*DD + xx];
        oacc1 += (M2[e1] + V1l[xx*4 + i1]) * xs[b1*DD + xx];
      }
      __syncthreads();
      for (int e = l; e < 64; e += 32) M2[e] = 0.f;
      __syncthreads();
    }

    // store out[node, c, i]  (layout (N, C, 4): 0e then 1o)
    {
      int e0 = l, e1 = l + 32;
      int i0 = e0 >> 4, b0 = e0 & 15;
      int i1 = e1 >> 4, b1 = e1 & 15;
      int bi0 = nt*16 + b0, bi1 = nt*16 + b1;
      if (bi0 < cnt) out[((size_t)nodelist[off + bi0]*CC + c)*4 + i0] = oacc0;
      if (bi1 < cnt) out[((size_t)nodelist[off + bi1]*CC + c)*4 + i1] = oacc1;
    }
    __syncthreads();
  }
}

// ---------------------------------------------------------------------------
extern "C" void kernel_launch(void* const* d_in, const int* in_sizes, int n_in,
                              void* d_out, int out_size, void* d_ws, size_t ws_size,
                              hipStream_t stream) {
  const float* x      = (const float*)d_in[0];
  const int*   index  = (const int*)  d_in[1];
  const float* st     = (const float*)d_in[2];
  const float* U3_0e  = (const float*)d_in[3];
  const float* U3_1o  = (const float*)d_in[4];
  const float* U2_0e  = (const float*)d_in[5];
  const float* U2_1o  = (const float*)d_in[6];
  const float* U1_0e  = (const float*)d_in[7];
  const float* U1_1o  = (const float*)d_in[8];
  const float* W3_0e  = (const float*)d_in[9];
  const float* W3_1o  = (const float*)d_in[10];
  const float* W2_0e  = (const float*)d_in[11];
  const float* W2_1o  = (const float*)d_in[12];
  const float* W1_0e  = (const float*)d_in[13];
  const float* W1_1o  = (const float*)d_in[14];
  float* out = (float*)d_out;

  // workspace layout (floats); ~8 MB of packed operands + 64 MB V3 = ~72 MB
  float* wsf    = (float*)d_ws;
  float* w3t0   = wsf;                       // 64*128*36 = 294912
  float* w3t1   = w3t0 + SS*CC*KP;           // 294912
  float* w2_0e  = w3t1 + SS*CC*KP;           // 32768
  float* w2_1o  = w2_0e + SS*4*CC;           // 49152
  float* w1_0e  = w2_1o + SS*6*CC;           // 8192
  float* w1_1o  = w1_0e + SS*CC;             // 8192
  float* UA     = w1_1o + SS*CC;             // 1024*16*36 = 589824
  int*   counts   = (int*)(UA + 1024*DD*KP); // 64
  int*   offsets  = counts + SS;             // 64
  int*   nodelist = offsets + SS;            // 1024
  float* V3     = wsf + (1u << 21);          // 8 MB in; 64 MB region (fits L2)

  int wtotal = SS*CC*KP*2 + SS*CC*(4+6+1+1);
  prep_weights<<<dim3((wtotal + 255)/256), 256, 0, stream>>>(
      st, W3_0e, W3_1o, W2_0e, W2_1o, W1_0e, W1_1o,
      w3t0, w3t1, w2_0e, w2_1o, w1_0e, w1_1o);

  prep_ua<<<dim3((1024*DD*KP + 255)/256), 256, 0, stream>>>(U3_0e, U3_1o, UA);

  bin_nodes<<<1, NN, 0, stream>>>(index, counts, offsets, nodelist);

  // 8 channel passes of 16 channels; V3 (64 MB) reused each pass, L2-resident
  for (int p = 0; p < 8; ++p) {
    build_v3<<<dim3(1024, SS), 32, 0, stream>>>(UA, w3t0, w3t1, V3, p*16);
    apply_contract<<<dim3(16, SS), 32, 0, stream>>>(
        x, V3, w2_0e, w2_1o, w1_0e, w1_1o,
        U2_0e, U2_1o, U1_0e, U1_1o,
        counts, offsets, nodelist, out, p*16);
  }
}